// TimesNetForecaster_61761629717047
// MI455X (gfx1250) — compile-verified
//
#include <hip/hip_runtime.h>
#include <math.h>

// ---- model constants ----
#define B_     8
#define SEQ_   96
#define PRED_  96
#define EXT_   192        // SEQ + PRED
#define D_     64
#define DFF_   64
#define NK_    6
#define TOPK_  5
#define MAXLEN_ 240       // max padded length: worst non-divisor period 38 -> 228
#define EPS_   1e-5f

typedef __attribute__((ext_vector_type(2))) float v2f;
typedef __attribute__((ext_vector_type(8))) float v8f;

// Full-precision fp32 WMMA: D(16x16 f32) = A(16x4 f32) * B(4x16 f32) + C
__device__ __forceinline__ v8f wmma_f32(v2f a, v2f b, v8f c) {
  // 8 args: (neg_a, A, neg_b, B, c_mod, C, reuse_a, reuse_b)
  return __builtin_amdgcn_wmma_f32_16x16x4_f32(false, a, false, b, (short)0, c,
                                               false, false);
}

struct IncW {
  const float* w[NK_];
  const float* b[NK_];
};

// ---------------------------------------------------------------------------
// Kernel 1: per-batch normalization + conv1d(k=3, wrap) embed + positional emb
// grid: (B_), block: (SEQ_)
// ---------------------------------------------------------------------------
__global__ void embed_kernel(const float* __restrict__ x,
                             const float* __restrict__ tw,
                             float* __restrict__ emb,
                             float* __restrict__ meanp,
                             float* __restrict__ stdp) {
  const int b = blockIdx.x;
  const int t = threadIdx.x;
  __shared__ float red[SEQ_];
  __shared__ float s_m, s_sd;

  const float xv = x[b * SEQ_ + t];
  red[t] = xv;
  __syncthreads();
  if (t == 0) {
    float s = 0.f;
    for (int i = 0; i < SEQ_; ++i) s += red[i];
    s_m = s / (float)SEQ_;
  }
  __syncthreads();
  const float m = s_m;
  red[t] = (xv - m) * (xv - m);
  __syncthreads();
  if (t == 0) {
    float s = 0.f;
    for (int i = 0; i < SEQ_; ++i) s += red[i];
    s_sd = sqrtf(s / (float)SEQ_ + EPS_);
    meanp[b] = m;
    stdp[b] = s_sd;
  }
  __syncthreads();
  const float inv_sd = 1.f / s_sd;

  const int tm = (t + SEQ_ - 1) % SEQ_;
  const int tp = (t + 1) % SEQ_;
  const float x0 = (x[b * SEQ_ + tm] - m) * inv_sd;
  const float x1 = (xv - m) * inv_sd;
  const float x2 = (x[b * SEQ_ + tp] - m) * inv_sd;

  const float neg_ln1e4_over_d = -9.210340371976184f / (float)D_;
  for (int d = 0; d < D_; ++d) {
    float acc = tw[d * 3 + 0] * x0 + tw[d * 3 + 1] * x1 + tw[d * 3 + 2] * x2;
    const int i2 = d >> 1;
    const float div = expf((float)(2 * i2) * neg_ln1e4_over_d);
    const float ang = (float)t * div;
    acc += (d & 1) ? cosf(ang) : sinf(ang);
    emb[((size_t)b * SEQ_ + t) * D_ + d] = acc;
  }
}

// ---------------------------------------------------------------------------
// Kernel 2: pred GEMM  cur[b,t,d] = sum_s pred_w[t,s] * emb[b,s,d] + pred_b[t]
// M=192, K=96, N=64 per batch.  fp32 WMMA 16x16x4.
// grid: (B_, 6), block: 256 (8 waves, one 16x16 tile per wave; 48 tiles/batch)
// ---------------------------------------------------------------------------
__global__ void pred_kernel(const float* __restrict__ emb,
                            const float* __restrict__ pw,
                            const float* __restrict__ pb,
                            float* __restrict__ cur) {
  const int b = blockIdx.x;
  const int lane = threadIdx.x & 31;
  const int wave = threadIdx.x >> 5;
  const int tileId = blockIdx.y * 8 + wave;       // 0..47 (uniform per wave)
  const int mt = tileId % 12;                     // t-tile
  const int nt = tileId / 12;                     // d-tile
  const int t0 = mt * 16;
  const int d0 = nt * 16;
  const int n = lane & 15;
  const int kh = lane >> 4;                       // 0/1 -> K rows {0,1}/{2,3}

  v8f acc = {};
  for (int s0 = 0; s0 < SEQ_; s0 += 4) {
    const int s = s0 + 2 * kh;
    v2f af, bf;
    af[0] = pw[(t0 + n) * SEQ_ + s];
    af[1] = pw[(t0 + n) * SEQ_ + s + 1];
    bf[0] = emb[((size_t)b * SEQ_ + s) * D_ + d0 + n];
    bf[1] = emb[((size_t)b * SEQ_ + s + 1) * D_ + d0 + n];
    acc = wmma_f32(af, bf, acc);
  }
#pragma unroll
  for (int r = 0; r < 8; ++r) {
    const int t = t0 + r + 8 * kh;
    cur[((size_t)b * EXT_ + t) * D_ + d0 + n] = acc[r] + pb[t];
  }
}

// ---------------------------------------------------------------------------
// Kernel 3: DFT amplitude (mean over batch & channels of |rfft|)
// grid: (97), block: 256
// ---------------------------------------------------------------------------
__global__ void amp_kernel(const float* __restrict__ cur,
                           float* __restrict__ amp) {
  const int f = blockIdx.x;                        // 0..96
  __shared__ float red[256];
  float local = 0.f;
  const float w0 = -2.f * 3.14159265358979f * (float)f / (float)EXT_;
  for (int pair = threadIdx.x; pair < B_ * D_; pair += 256) {
    const int b = pair >> 6;
    const int d = pair & 63;
    float re = 0.f, im = 0.f;
    const float* xp = cur + (size_t)b * EXT_ * D_ + d;
    for (int t = 0; t < EXT_; ++t) {
      float s, c;
      sincosf(w0 * (float)t, &s, &c);
      const float v = xp[t * D_];
      re += v * c;
      im += v * s;
    }
    local += sqrtf(re * re + im * im);
  }
  red[threadIdx.x] = local;
  __syncthreads();
  for (int s = 128; s > 0; s >>= 1) {
    if (threadIdx.x < s) red[threadIdx.x] += red[threadIdx.x + s];
    __syncthreads();
  }
  if (threadIdx.x == 0) amp[f] = red[0] / (float)(B_ * D_);
}

// ---------------------------------------------------------------------------
// Kernel 4: top-5 period selection (serial, tiny)
// sel layout (ints): [0..4]=idx, [5..9]=period, [10..14]=length, [15..19]=H
// ---------------------------------------------------------------------------
__global__ void topk_kernel(const float* __restrict__ amp,
                            int* __restrict__ sel) {
  if (threadIdx.x != 0) return;
  float a[97];
  for (int i = 0; i < 97; ++i) a[i] = amp[i];
  a[0] = -1.f;
  for (int j = 0; j < TOPK_; ++j) {
    int best = 1;
    float bv = a[1];
    for (int i = 2; i < 97; ++i)
      if (a[i] > bv) { bv = a[i]; best = i; }    // strict > == stable argsort
    a[best] = -1.f;
    const int p = EXT_ / best;
    const int len = ((EXT_ + p - 1) / p) * p;
    sel[j] = best;
    sel[TOPK_ + j] = p;
    sel[2 * TOPK_ + j] = len;
    sel[3 * TOPK_ + j] = len / p;
  }
}

// ---------------------------------------------------------------------------
// Kernel 5: inception conv stack (mean of 6 SAME conv2d, kernels 1..11) via
// fp32 WMMA over the channel dimension. One block per (batch, period-slot);
// full activation grid staged in LDS (64 ch x MAXLEN <= 60KB).
// PASS==1: input cur[b,t,c], output s1[slot][b][ci][pos] = gelu(inception1)
// PASS==2: input s1,        output s2[slot][b][t][c]    = inception2 (t<EXT)
// grid: (B_, TOPK_), block: 256
// ---------------------------------------------------------------------------
template <int PASS>
__global__ void inception_kernel(const float* __restrict__ in,
                                 float* __restrict__ out,
                                 IncW wts,
                                 const int* __restrict__ sel) {
  __shared__ float lds[D_ * MAXLEN_];
  const int b = blockIdx.x;
  const int slot = blockIdx.y;
  const int p = sel[TOPK_ + slot];
  const int length = sel[2 * TOPK_ + slot];
  const int H = sel[3 * TOPK_ + slot];

  // ---- stage input grid into LDS as [ci][pos] ----
  const float* src = (PASS == 1)
                         ? (in + (size_t)b * EXT_ * D_)
                         : (in + ((size_t)slot * B_ + b) * (D_ * MAXLEN_));
  for (int i = threadIdx.x; i < D_ * MAXLEN_; i += 256) {
    const int ci = i / MAXLEN_;
    const int pos = i - ci * MAXLEN_;
    float v = 0.f;
    if (pos < length) {
      if (PASS == 1) {
        if (pos < EXT_) v = src[(size_t)pos * D_ + ci];   // pad beyond EXT = 0
      } else {
        v = src[(size_t)ci * MAXLEN_ + pos];
      }
    }
    lds[i] = v;
  }
  __syncthreads();

  const int lane = threadIdx.x & 31;
  const int wave = threadIdx.x >> 5;
  const int n = lane & 15;                 // A row / B col / C col
  const int kh = lane >> 4;                // K-half: rows {0,1} or {2,3}
  const int nPT = (length + 15) >> 4;
  const int nTiles = 4 * nPT;              // 4 f-tiles x pos-tiles

  for (int tile = wave; tile < nTiles; tile += 8) {
    const int ft = tile & 3;
    const int pt = tile >> 2;
    const int f0 = ft * 16;
    const int pos0 = pt * 16;
    const int pos = pos0 + n;
    const int cpos = (pos < length) ? pos : (length - 1);  // keep EXEC full
    const int hh = cpos / p;
    const int ww = cpos - hh * p;

    v8f acc = {};
#pragma unroll 1
    for (int k = 0; k < NK_; ++k) {
      const int ksz = 2 * k + 1;
      const int kk2 = ksz * ksz;
      const float* wp = wts.w[k];
#pragma unroll 1
      for (int dy = 0; dy < ksz; ++dy) {
        const int oy = dy - k;
        const bool vy = ((unsigned)(hh + oy) < (unsigned)H);
#pragma unroll 1
        for (int dx = 0; dx < ksz; ++dx) {
          const int ox = dx - k;
          const bool valid = vy && ((unsigned)(ww + ox) < (unsigned)p);
          const int spos = cpos + oy * p + ox;             // in [0,length) if valid
          const int tap = dy * ksz + dx;
          const float* wA = wp + (size_t)(f0 + n) * D_ * kk2 + tap;
          for (int ci0 = 0; ci0 < D_; ci0 += 4) {
            const int ci = ci0 + 2 * kh;
            v2f bf, af;
            bf[0] = valid ? lds[ci * MAXLEN_ + spos] : 0.f;
            bf[1] = valid ? lds[(ci + 1) * MAXLEN_ + spos] : 0.f;
            af[0] = wA[(size_t)ci * kk2];
            af[1] = wA[(size_t)(ci + 1) * kk2];
            acc = wmma_f32(af, bf, acc);
          }
        }
      }
    }

    // ---- epilogue: + mean bias, (gelu), store ----
#pragma unroll
    for (int r = 0; r < 8; ++r) {
      const int f = f0 + r + 8 * kh;
      float bs = 0.f;
#pragma unroll
      for (int k = 0; k < NK_; ++k) bs += wts.b[k][f];
      float v = (acc[r] + bs) * (1.f / 6.f);
      if (PASS == 1) {
        v = 0.5f * v * (1.f + erff(v * 0.70710678118654752f));  // exact gelu
        if (pos < length)
          out[((size_t)slot * B_ + b) * (D_ * MAXLEN_) + (size_t)f * MAXLEN_ + pos] = v;
      } else {
        if (pos < EXT_)
          out[(((size_t)slot * B_ + b) * EXT_ + pos) * D_ + f] = v;
      }
    }
  }
}

// ---------------------------------------------------------------------------
// Kernel 6: agg(sum of 5 slots) + residual + layernorm  (period weights == 1)
// grid: (EXT_, B_), block: D_
// ---------------------------------------------------------------------------
__global__ void resln_kernel(const float* __restrict__ cur,
                             const float* __restrict__ s2,
                             const float* __restrict__ g,
                             const float* __restrict__ bb,
                             float* __restrict__ outc) {
  const int t = blockIdx.x;
  const int b = blockIdx.y;
  const int c = threadIdx.x;
  __shared__ float red[D_];

  float v = cur[((size_t)b * EXT_ + t) * D_ + c];
  for (int s = 0; s < TOPK_; ++s)
    v += s2[(((size_t)s * B_ + b) * EXT_ + t) * D_ + c];

  red[c] = v;
  __syncthreads();
  for (int st = 32; st > 0; st >>= 1) {
    if (c < st) red[c] += red[c + st];
    __syncthreads();
  }
  const float m = red[0] / (float)D_;
  __syncthreads();
  red[c] = (v - m) * (v - m);
  __syncthreads();
  for (int st = 32; st > 0; st >>= 1) {
    if (c < st) red[c] += red[c + st];
    __syncthreads();
  }
  const float var = red[0] / (float)D_;
  outc[((size_t)b * EXT_ + t) * D_ + c] =
      (v - m) * rsqrtf(var + EPS_) * g[c] + bb[c];
}

// ---------------------------------------------------------------------------
// Kernel 7: final projection + de-normalization
// grid: (B_), block: (PRED_)
// ---------------------------------------------------------------------------
__global__ void final_kernel(const float* __restrict__ cur,
                             const float* __restrict__ pw,
                             const float* __restrict__ pb,
                             const float* __restrict__ meanp,
                             const float* __restrict__ stdp,
                             float* __restrict__ out) {
  const int b = blockIdx.x;
  const int i = threadIdx.x;
  const int t = EXT_ - PRED_ + i;
  float acc = pb[0];
  const float* row = cur + ((size_t)b * EXT_ + t) * D_;
  for (int c = 0; c < D_; ++c) acc += row[c] * pw[c];
  out[b * PRED_ + i] = acc * stdp[b] + meanp[b];
}

// ---------------------------------------------------------------------------
extern "C" void kernel_launch(void* const* d_in, const int* in_sizes, int n_in,
                              void* d_out, int out_size, void* d_ws,
                              size_t ws_size, hipStream_t stream) {
  (void)in_sizes; (void)n_in; (void)out_size; (void)ws_size;

  const float* x       = (const float*)d_in[0];
  const float* token_w = (const float*)d_in[1];
  const float* pred_w  = (const float*)d_in[2];
  const float* pred_b  = (const float*)d_in[3];
  const int PER_BLK = 26;  // 6 inc1_w + 6 inc1_b + 6 inc2_w + 6 inc2_b + ln_g + ln_b
  const float* proj_w = (const float*)d_in[4 + 2 * PER_BLK];
  const float* proj_b = (const float*)d_in[4 + 2 * PER_BLK + 1];

  // workspace layout (floats)
  float* ws      = (float*)d_ws;
  float* w_mean  = ws + 0;                           // 8
  float* w_std   = ws + 8;                           // 8
  float* w_emb   = ws + 64;                          // 8*96*64   = 49152
  float* w_curA  = w_emb + (size_t)B_ * SEQ_ * D_;   // 8*192*64  = 98304
  float* w_curB  = w_curA + (size_t)B_ * EXT_ * D_;  // 98304
  float* w_amp   = w_curB + (size_t)B_ * EXT_ * D_;  // 128
  int*   w_sel   = (int*)(w_amp + 128);              // 32 ints
  float* w_s1    = w_amp + 160;                      // 5*8*64*240 = 614400
  float* w_s2    = w_s1 + (size_t)TOPK_ * B_ * D_ * MAXLEN_;  // 5*8*192*64 = 491520

  float* cur = w_curA;
  float* nxt = w_curB;

  embed_kernel<<<B_, SEQ_, 0, stream>>>(x, token_w, w_emb, w_mean, w_std);
  pred_kernel<<<dim3(B_, 6), 256, 0, stream>>>(w_emb, pred_w, pred_b, cur);

  for (int layer = 0; layer < 2; ++layer) {
    void* const* bi = d_in + 4 + layer * PER_BLK;
    IncW w1, w2;
    for (int k = 0; k < NK_; ++k) {
      w1.w[k] = (const float*)bi[k];
      w1.b[k] = (const float*)bi[6 + k];
      w2.w[k] = (const float*)bi[12 + k];
      w2.b[k] = (const float*)bi[18 + k];
    }
    const float* ln_g = (const float*)bi[24];
    const float* ln_b = (const float*)bi[25];

    amp_kernel<<<EXT_ / 2 + 1, 256, 0, stream>>>(cur, w_amp);
    topk_kernel<<<1, 32, 0, stream>>>(w_amp, w_sel);
    inception_kernel<1><<<dim3(B_, TOPK_), 256, 0, stream>>>(cur, w_s1, w1, w_sel);
    inception_kernel<2><<<dim3(B_, TOPK_), 256, 0, stream>>>(w_s1, w_s2, w2, w_sel);
    resln_kernel<<<dim3(EXT_, B_), D_, 0, stream>>>(cur, w_s2, ln_g, ln_b, nxt);

    float* tmp = cur; cur = nxt; nxt = tmp;
  }

  final_kernel<<<B_, PRED_, 0, stream>>>(cur, proj_w, proj_b, w_mean, w_std,
                                         (float*)d_out);
}